// Loss_yolov1_65017214927610
// MI455X (gfx1250) — compile-verified
//
#include <hip/hip_runtime.h>

#define G 7
#define CELLS 49          // G*G
#define CH 30
#define IMG_STRIDE (CH * CELLS)   // 1470 floats per image

typedef __attribute__((ext_vector_type(2))) float v2f;
typedef __attribute__((ext_vector_type(8))) float v8f;

// Full wave32 sum using V_WMMA_F32_16X16X4_F32.
// A (16x4 f32, 2 VGPRs): lanes 0-15 -> {K0, K1}, lanes 16-31 -> {K2, K3}, M = lane%16.
// Put s in VGPR0, 0 in VGPR1; B = all ones  =>  D[m,n] = s_m + s_{m+16} (any n).
// D VGPR r at lane L holds D[r + (L<16?0:8), L%16]; summing the 8 VGPRs gives
// lanes 0-15: sum of rows 0-7, lanes 16-31: sum of rows 8-15. One xor-16 shuffle
// then yields the full 32-lane sum in every lane.
__device__ __forceinline__ float wave_sum32(float s) {
    v2f a; a[0] = s;    a[1] = 0.0f;
    v2f b; b[0] = 1.0f; b[1] = 1.0f;
    v8f c = {};
    c = __builtin_amdgcn_wmma_f32_16x16x4_f32(
            /*neg_a=*/false, a, /*neg_b=*/false, b,
            /*c_mod=*/(short)0, c, /*reuse_a=*/false, /*reuse_b=*/false);
    float t = c[0] + c[1] + c[2] + c[3] + c[4] + c[5] + c[6] + c[7];
    t += __shfl_xor(t, 16, 32);
    return t;
}

__device__ __forceinline__ float iou_box(float ax1, float ay1, float ax2, float ay2,
                                         float bx1, float by1, float bx2, float by2) {
    float ix1 = fmaxf(ax1, bx1), iy1 = fmaxf(ay1, by1);
    float ix2 = fminf(ax2, bx2), iy2 = fminf(ay2, by2);
    float iw = fmaxf(ix2 - ix1, 0.0f), ih = fmaxf(iy2 - iy1, 0.0f);
    float inter = iw * ih;
    float aa = (ax2 - ax1) * (ay2 - ay1);
    float ab = (bx2 - bx1) * (by2 - by1);
    return inter > 0.0f ? inter / (aa + ab - inter) : 0.0f;
}

__global__ void __launch_bounds__(256)
yolo_loss_partial(const float* __restrict__ pred,
                  const float* __restrict__ labels,
                  float* __restrict__ partial,
                  int total_cells) {
    float s = 0.0f;
    const int stride = gridDim.x * blockDim.x;
    for (int i = blockIdx.x * blockDim.x + threadIdx.x; i < total_cells; i += stride) {
        const int img = i / CELLS;
        const int p   = i - img * CELLS;        // p = m*G + n
        const float* pb = pred   + (size_t)img * IMG_STRIDE + p;
        const float* lb = labels + (size_t)img * IMG_STRIDE + p;

        float pv[CH], lv[CH];
#pragma unroll
        for (int c = 0; c < CH; ++c) pv[c] = pb[c * CELLS];
#pragma unroll
        for (int c = 0; c < CH; ++c) lv[c] = lb[c * CELLS];

        const int   mi = p / G;
        const float mf = (float)mi;
        const float nf = (float)(p - mi * G);

        // box1 (pred 0..3), box2 (pred 5..8), gt (labels 0..3) in xyxy
        float b1x1, b1y1, b1x2, b1y2, b2x1, b2y1, b2x2, b2y2, gx1, gy1, gx2, gy2;
        {
            float cx = (pv[0] + mf) / 7.0f, cy = (pv[1] + nf) / 7.0f;
            float w = pv[2], h = pv[3];
            b1x1 = cx - 0.5f * w; b1y1 = cy - 0.5f * h;
            b1x2 = cx + 0.5f * w; b1y2 = cy + 0.5f * h;
        }
        {
            float cx = (pv[5] + mf) / 7.0f, cy = (pv[6] + nf) / 7.0f;
            float w = pv[7], h = pv[8];
            b2x1 = cx - 0.5f * w; b2y1 = cy - 0.5f * h;
            b2x2 = cx + 0.5f * w; b2y2 = cy + 0.5f * h;
        }
        {
            float cx = (lv[0] + mf) / 7.0f, cy = (lv[1] + nf) / 7.0f;
            float w = lv[2], h = lv[3];
            gx1 = cx - 0.5f * w; gy1 = cy - 0.5f * h;
            gx2 = cx + 0.5f * w; gy2 = cy + 0.5f * h;
        }

        const float iou1 = iou_box(b1x1, b1y1, b1x2, b1y2, gx1, gy1, gx2, gy2);
        const float iou2 = iou_box(b2x1, b2y1, b2x2, b2y2, gx1, gy1, gx2, gy2);
        const bool use1 = (iou1 >= iou2);
        const bool obj  = (lv[4] == 1.0f);

        float d0 = pv[0] - lv[0], d1 = pv[1] - lv[1];
        float s2 = sqrtf(pv[2]) - sqrtf(lv[2]);
        float s3 = sqrtf(pv[3]) - sqrtf(lv[3]);
        float coor1 = d0 * d0 + d1 * d1 + s2 * s2 + s3 * s3;

        float d5 = pv[5] - lv[5], d6 = pv[6] - lv[6];
        float s7 = sqrtf(pv[7]) - sqrtf(lv[7]);
        float s8 = sqrtf(pv[8]) - sqrtf(lv[8]);
        float coor2 = d5 * d5 + d6 * d6 + s7 * s7 + s8 * s8;

        float coor = 5.0f * (use1 ? coor1 : coor2);

        float e1 = pv[4] - iou1; e1 *= e1;
        float e2 = pv[9] - iou2; e2 *= e2;
        float obj_conf          = use1 ? e1 : e2;
        float noobj_conf_objc   = 0.5f * (use1 ? e2 : e1);
        float noobj_conf_nocell = 0.5f * (pv[4] * pv[4] + pv[9] * pv[9]);

        float cls = 0.0f;
#pragma unroll
        for (int c = 10; c < CH; ++c) {
            float d = pv[c] - lv[c];
            cls += d * d;
        }

        s += obj ? (coor + obj_conf + noobj_conf_objc + cls) : noobj_conf_nocell;
    }

    // All threads (in-range or not) reach here with full EXEC -> WMMA legal.
    float t = wave_sum32(s);

    __shared__ float wsum[8];
    const int lane = threadIdx.x & 31;
    const int wave = threadIdx.x >> 5;
    if (lane == 0) wsum[wave] = t;
    __syncthreads();
    if (threadIdx.x == 0) {
        float bs = 0.0f;
        const int nw = blockDim.x >> 5;
        for (int w = 0; w < nw; ++w) bs += wsum[w];
        partial[blockIdx.x] = bs;
    }
}

__global__ void __launch_bounds__(256)
yolo_loss_finalize(const float* __restrict__ partial, float* __restrict__ out,
                   int nparts, float inv_b) {
    __shared__ float sm[256];
    float s = 0.0f;
    for (int j = threadIdx.x; j < nparts; j += 256) s += partial[j];
    sm[threadIdx.x] = s;
    __syncthreads();
    for (int k = 128; k > 0; k >>= 1) {
        if (threadIdx.x < (unsigned)k) sm[threadIdx.x] += sm[threadIdx.x + k];
        __syncthreads();
    }
    if (threadIdx.x == 0) out[0] = sm[0] * inv_b;
}

extern "C" void kernel_launch(void* const* d_in, const int* in_sizes, int n_in,
                              void* d_out, int out_size, void* d_ws, size_t ws_size,
                              hipStream_t stream) {
    const float* pred   = (const float*)d_in[0];
    const float* labels = (const float*)d_in[1];
    float* out     = (float*)d_out;
    float* partial = (float*)d_ws;

    const int B     = in_sizes[0] / IMG_STRIDE;   // 8192
    const int total = B * CELLS;                  // 401408 cells
    const int threads = 256;

    int blocks = (total + threads - 1) / threads; // 1568 for B=8192
    int maxb = (int)(ws_size / sizeof(float));
    if (maxb > 2048) maxb = 2048;
    if (maxb < 1)    maxb = 1;
    if (blocks > maxb) blocks = maxb;             // grid-stride covers the rest

    yolo_loss_partial<<<blocks, threads, 0, stream>>>(pred, labels, partial, total);
    yolo_loss_finalize<<<1, 256, 0, stream>>>(partial, out, blocks, 1.0f / (float)B);
}